// ModelEuCoHM_73005854097656
// MI455X (gfx1250) — compile-verified
//
#include <hip/hip_runtime.h>

#define HID 128
#define NEG_SLOPE 0.2f
#define BN_EPS 1e-5f

typedef float v2f __attribute__((ext_vector_type(2)));
typedef float v8f __attribute__((ext_vector_type(8)));
typedef unsigned u32x4 __attribute__((ext_vector_type(4)));
typedef int i32x4 __attribute__((ext_vector_type(4)));
typedef int i32x8 __attribute__((ext_vector_type(8)));

__device__ __forceinline__ float lrelu(float x) { return x > 0.f ? x : NEG_SLOPE * x; }

// ================= dense transform: C[N,128] = A[N,128] @ W[128,128] =================
// One wave per 16x16 C tile, fp32 WMMA (V_WMMA_F32_16X16X4_F32), K stepped by 4.
// The block's shared 16x128 A tile (8KB) is staged into LDS by a single TDM
// tensor_load_to_lds issued by wave 0 (D# built per CDNA5 ISA),
// synchronized with s_wait_tensorcnt + workgroup barrier.
__global__ void gemm_n128(const float* __restrict__ A, const float* __restrict__ W,
                          float* __restrict__ C, int nrows) {
    (void)nrows; // 50000 % 16 == 0, no tail
    __shared__ float As[16 * HID];

    const int m0 = blockIdx.x * 16;
    if (threadIdx.x < 32) {
        unsigned lds = (unsigned)(size_t)(&As[0]);
        unsigned long long ga = (unsigned long long)(size_t)(A + (size_t)m0 * HID);
        u32x4 g0;
        g0.x = 1u;                                   // count=1 (valid user descriptor)
        g0.y = lds;                                  // lds_addr (bytes)
        g0.z = (unsigned)(ga & 0xffffffffu);         // global_addr[31:0]
        g0.w = (unsigned)((ga >> 32) & 0x01ffffffu)  // global_addr[56:32]
             | (2u << 30);                           // type = 2 ("image")
        i32x8 g1;
        g1[0] = 0x20000;              // workgroup_mask=0, data_size=2 (4 bytes)
        g1[1] = (int)(128u << 16);    // tensor_dim0 = 128 elements
        g1[2] = (int)(16u << 16);     // tensor_dim1 = 16 rows (tile-local tensor)
        g1[3] = (int)(128u << 16);    // tile_dim0 = 128
        g1[4] = 16;                   // tile_dim1 = 16, tile_dim2 = 0
        g1[5] = 128;                  // tensor_dim0_stride = 128 elements
        g1[6] = 0;
        g1[7] = 0;
        i32x4 gz4 = {0, 0, 0, 0};
        i32x8 gz8 = {0, 0, 0, 0, 0, 0, 0, 0};
        __builtin_amdgcn_tensor_load_to_lds(g0, g1, gz4, gz4, gz8, 0);
        __builtin_amdgcn_s_wait_tensorcnt(0);
    }
    __syncthreads();

    const int lane  = threadIdx.x & 31;
    const int wave  = threadIdx.x >> 5;          // 8 waves -> 8 N-tiles (128 cols)
    const int n0    = wave * 16;
    const int kshft = (lane >> 4) << 1;
    const int colB  = n0 + (lane & 15);
    const float* arow = &As[(lane & 15) * HID];

    v8f acc = {};
#pragma unroll
    for (int k0 = 0; k0 < HID; k0 += 4) {
        v2f a, b;
        a.x = arow[k0 + kshft];
        a.y = arow[k0 + kshft + 1];
        b.x = W[(size_t)(k0 + kshft) * HID + colB];
        b.y = W[(size_t)(k0 + kshft + 1) * HID + colB];
        acc = __builtin_amdgcn_wmma_f32_16x16x4_f32(false, a, false, b,
                                                    (short)0, acc, false, false);
    }
    const int rbase = m0 + ((lane >> 4) << 3);
    const int col   = n0 + (lane & 15);
#pragma unroll
    for (int r = 0; r < 8; ++r)
        C[(size_t)(rbase + r) * HID + col] = acc[r];
}

// ================= init helper =================
__global__ void fill_u32(unsigned* __restrict__ p, unsigned v, int n) {
    for (int i = blockIdx.x * blockDim.x + threadIdx.x; i < n; i += gridDim.x * blockDim.x)
        p[i] = v;
}

// ================= CSR construction (dst is layer-invariant: build once) =============
__global__ void count_deg(const int* __restrict__ ei, unsigned* __restrict__ deg,
                          int ne, int etot) {
    int e = blockIdx.x * blockDim.x + threadIdx.x;
    if (e >= etot) return;
    int d = (e < ne) ? ei[ne + e] : (e - ne);
    atomicAdd(&deg[d], 1u);
}

// single-block exclusive scan: rowstart[0..n] from deg[0..n-1]
__global__ void scan_deg(const unsigned* __restrict__ deg, unsigned* __restrict__ rowstart,
                         int n) {
    __shared__ unsigned tmp[1024];
    __shared__ unsigned carry;
    if (threadIdx.x == 0) carry = 0;
    __syncthreads();
    for (int base = 0; base < n; base += 1024) {
        int i = base + (int)threadIdx.x;
        unsigned v = (i < n) ? deg[i] : 0u;
        tmp[threadIdx.x] = v;
        __syncthreads();
        for (int o = 1; o < 1024; o <<= 1) {            // Hillis-Steele inclusive scan
            unsigned t = (threadIdx.x >= (unsigned)o) ? tmp[threadIdx.x - o] : 0u;
            __syncthreads();
            tmp[threadIdx.x] += t;
            __syncthreads();
        }
        if (i < n) rowstart[i] = carry + tmp[threadIdx.x] - v;
        __syncthreads();
        if (threadIdx.x == 1023) carry += tmp[1023];
        __syncthreads();
    }
    if (threadIdx.x == 0) rowstart[n] = carry;
}

__global__ void csr_scatter(const int* __restrict__ ei, const unsigned* __restrict__ rowstart,
                            unsigned* __restrict__ cursor, unsigned* __restrict__ csr,
                            int ne, int etot) {
    int e = blockIdx.x * blockDim.x + threadIdx.x;
    if (e >= etot) return;
    int d = (e < ne) ? ei[ne + e] : (e - ne);
    unsigned pos = atomicAdd(&cursor[d], 1u);
    csr[rowstart[d] + pos] = (unsigned)e;
}

// ================= edge pass: e = att . lrelu(xl[src]+xr[dst]) =================
__global__ void edge_score(const float4* __restrict__ xl4, const float4* __restrict__ xr4,
                           const float4* __restrict__ att4, const int* __restrict__ ei,
                           int ne, int etot, float* __restrict__ esc) {
    int e    = (int)((blockIdx.x * blockDim.x + threadIdx.x) >> 5);
    int lane = threadIdx.x & 31;
    if (e >= etot) return;
    int s, d;
    if (e < ne) { s = ei[e]; d = ei[ne + e]; } else { s = d = e - ne; }
    float4 va = xl4[(size_t)s * 32 + lane];
    float4 vb = xr4[(size_t)d * 32 + lane];
    float4 aa = att4[lane];
    float p = aa.x * lrelu(va.x + vb.x) + aa.y * lrelu(va.y + vb.y)
            + aa.z * lrelu(va.z + vb.z) + aa.w * lrelu(va.w + vb.w);
#pragma unroll
    for (int o = 16; o > 0; o >>= 1) p += __shfl_xor(p, o, 32);
    if (lane == 0) esc[e] = p;
}

// ============ per-dst softmax over incoming edges (no atomics, wave reductions) =======
__global__ void dst_softmax(float* __restrict__ esc, const unsigned* __restrict__ rowstart,
                            const unsigned* __restrict__ csr, int n) {
    int d    = (int)((blockIdx.x * blockDim.x + threadIdx.x) >> 5);
    int lane = threadIdx.x & 31;
    if (d >= n) return;
    unsigned s0 = rowstart[d], s1 = rowstart[d + 1];
    float m = -3.4e38f;
    for (unsigned i = s0 + lane; i < s1; i += 32) m = fmaxf(m, esc[csr[i]]);
#pragma unroll
    for (int o = 16; o > 0; o >>= 1) m = fmaxf(m, __shfl_xor(m, o, 32));
    float s = 0.f;
    for (unsigned i = s0 + lane; i < s1; i += 32) s += expf(esc[csr[i]] - m);
#pragma unroll
    for (int o = 16; o > 0; o >>= 1) s += __shfl_xor(s, o, 32);
    float inv = 1.f / s;
    for (unsigned i = s0 + lane; i < s1; i += 32) {
        unsigned e = csr[i];
        esc[e] = expf(esc[e] - m) * inv;     // esc becomes alpha
    }
}

// ============ per-dst aggregation: out[d] = bias + sum alpha[e] * xl[src[e]] ==========
// One wave per dst row; float4/lane covers 128 features; single coalesced write.
__global__ void dst_aggregate(const float* __restrict__ esc, const float4* __restrict__ xl4,
                              const int* __restrict__ ei, const unsigned* __restrict__ rowstart,
                              const unsigned* __restrict__ csr, const float4* __restrict__ bias4,
                              float4* __restrict__ out4, int ne, int n) {
    int d    = (int)((blockIdx.x * blockDim.x + threadIdx.x) >> 5);
    int lane = threadIdx.x & 31;
    if (d >= n) return;
    float4 acc = bias4[lane];
    unsigned s1 = rowstart[d + 1];
    for (unsigned i = rowstart[d]; i < s1; ++i) {
        unsigned e = csr[i];
        int s = (e < (unsigned)ne) ? ei[e] : (int)(e - (unsigned)ne);
        float a = esc[e];
        float4 v = xl4[(size_t)s * 32 + lane];
        acc.x += a * v.x; acc.y += a * v.y; acc.z += a * v.z; acc.w += a * v.w;
    }
    out4[(size_t)d * 32 + lane] = acc;
}

// ================= batchnorm =================
// grid*block is a multiple of 128 so each thread stays in one column.
__global__ void bn_stats(const float* __restrict__ h, float* __restrict__ sums,
                         float* __restrict__ sumsq, int n) {
    int stride = gridDim.x * blockDim.x;
    int t = blockIdx.x * blockDim.x + threadIdx.x;
    float s = 0.f, s2 = 0.f;
    for (int i = t; i < n; i += stride) { float v = h[i]; s += v; s2 += v * v; }
    atomicAdd(&sums[t & (HID - 1)], s);
    atomicAdd(&sumsq[t & (HID - 1)], s2);
}

__global__ void bn_apply(const float* __restrict__ hb, const float* __restrict__ sums,
                         const float* __restrict__ sumsq, const float* __restrict__ gamma,
                         const float* __restrict__ beta, float* __restrict__ hout,
                         float* __restrict__ acc, float invN, float resw, int n) {
    int t = blockIdx.x * blockDim.x + threadIdx.x;
    if (t >= n) return;
    int c = t & (HID - 1);
    float mu  = sums[c] * invN;
    float var = sumsq[c] * invN - mu * mu;
    float y = (hb[t] - mu) * rsqrtf(var + BN_EPS) * gamma[c] + beta[c];
    hout[t] = y;
    acc[t] += y * resw;
}

// ================= final: dot(out[u], out[v]) per labeled pair =================
__global__ void pair_dot(const float4* __restrict__ acc4, const int* __restrict__ eli,
                         float* __restrict__ out, int np) {
    int p    = (int)((blockIdx.x * blockDim.x + threadIdx.x) >> 5);
    int lane = threadIdx.x & 31;
    if (p >= np) return;
    int u = eli[p], v = eli[np + p];
    float4 a = acc4[(size_t)u * 32 + lane];
    float4 b = acc4[(size_t)v * 32 + lane];
    float s = a.x * b.x + a.y * b.y + a.z * b.z + a.w * b.w;
#pragma unroll
    for (int o = 16; o > 0; o >>= 1) s += __shfl_xor(s, o, 32);
    if (lane == 0) out[p] = s;
}

extern "C" void kernel_launch(void* const* d_in, const int* in_sizes, int n_in,
                              void* d_out, int out_size, void* d_ws, size_t ws_size,
                              hipStream_t stream) {
    (void)n_in; (void)out_size; (void)ws_size;
    const float* x     = (const float*)d_in[0];
    const int*   ei    = (const int*)d_in[1];
    const int*   eli   = (const int*)d_in[2];
    const float* Wl    = (const float*)d_in[3];
    const float* Wr    = (const float*)d_in[4];
    const float* att   = (const float*)d_in[5];
    const float* bias  = (const float*)d_in[6];
    const float* gamma = (const float*)d_in[7];
    const float* beta  = (const float*)d_in[8];
    float* out = (float*)d_out;

    const int N  = in_sizes[0] / HID;   // 50000
    const int NE = in_sizes[1] / 2;     // 1,600,000
    const int NP = in_sizes[2] / 2;     // 8192
    const int ET = NE + N;              // edges + self loops
    const int NH = N * HID;

    char* w = (char*)d_ws;
    float*    hbuf     = (float*)w;    w += (size_t)NH * 4;
    float*    xl       = (float*)w;    w += (size_t)NH * 4;
    float*    xr       = (float*)w;    w += (size_t)NH * 4;
    float*    outbuf   = (float*)w;    w += (size_t)NH * 4;
    float*    acc      = (float*)w;    w += (size_t)NH * 4;
    float*    esc      = (float*)w;    w += (size_t)ET * 4;
    unsigned* csr      = (unsigned*)w; w += (size_t)ET * 4;
    unsigned* rowstart = (unsigned*)w; w += (size_t)(N + 1) * 4;
    unsigned* deg      = (unsigned*)w; w += (size_t)N * 4;     // reused as scatter cursor
    float*    sums     = (float*)w;    w += (size_t)HID * 4;
    float*    sumsq    = (float*)w;

    const int B = 256;
    const unsigned gEdgeW = (unsigned)(((size_t)ET * 32 + B - 1) / B);
    const unsigned gEdge  = (unsigned)((ET + B - 1) / B);
    const unsigned gDstW  = (unsigned)(((size_t)N * 32 + B - 1) / B);
    const unsigned gElem  = (unsigned)((NH + B - 1) / B);

    // ---- residual accumulator + CSR (dst indices are layer-invariant) ----
    fill_u32<<<256, B, 0, stream>>>((unsigned*)acc, 0u, NH);
    fill_u32<<<64, B, 0, stream>>>(deg, 0u, N);
    count_deg<<<gEdge, B, 0, stream>>>(ei, deg, NE, ET);
    scan_deg<<<1, 1024, 0, stream>>>(deg, rowstart, N);
    fill_u32<<<64, B, 0, stream>>>(deg, 0u, N);                // becomes cursor
    csr_scatter<<<gEdge, B, 0, stream>>>(ei, rowstart, deg, csr, NE, ET);

    for (int L = 0; L < 4; ++L) {
        const float* hin = (L == 0) ? x : hbuf;
        gemm_n128<<<N / 16, B, 0, stream>>>(hin, Wl + (size_t)L * HID * HID, xl, N);
        gemm_n128<<<N / 16, B, 0, stream>>>(hin, Wr + (size_t)L * HID * HID, xr, N);

        edge_score<<<gEdgeW, B, 0, stream>>>((const float4*)xl, (const float4*)xr,
                                             (const float4*)(att + (size_t)L * HID),
                                             ei, NE, ET, esc);
        dst_softmax<<<gDstW, B, 0, stream>>>(esc, rowstart, csr, N);
        dst_aggregate<<<gDstW, B, 0, stream>>>(esc, (const float4*)xl, ei, rowstart, csr,
                                               (const float4*)(bias + (size_t)L * HID),
                                               (float4*)outbuf, NE, N);

        fill_u32<<<1, B, 0, stream>>>((unsigned*)sums, 0u, 2 * HID);
        bn_stats<<<512, B, 0, stream>>>(outbuf, sums, sumsq, NH);
        bn_apply<<<gElem, B, 0, stream>>>(outbuf, sums, sumsq,
                                          gamma + (size_t)L * HID, beta + (size_t)L * HID,
                                          hbuf, acc, 1.f / (float)N, 0.2f, NH);
    }

    pair_dot<<<(unsigned)(((size_t)NP * 32 + B - 1) / B), B, 0, stream>>>(
        (const float4*)acc, eli, out, NP);
}